// SPR_GNN_88648124990003
// MI455X (gfx1250) — compile-verified
//
#include <hip/hip_runtime.h>
#include <hip/hip_bf16.h>

#define N_NODES  100000
#define N_EDGES  1600000
#define N_GRAPHS 2048
#define EMBED    64
#define HID      64
#define N_CL     2

typedef __attribute__((ext_vector_type(2))) float v2f;
typedef __attribute__((ext_vector_type(8))) float v8f;

static __host__ int cdiv(int a, int b) { return (a + b - 1) / b; }

// ---------------- utility kernels ----------------

__global__ void k_fill_f32(float* __restrict__ p, float v, int n) {
    int i = blockIdx.x * blockDim.x + threadIdx.x;
    if (i < n) p[i] = v;
}

__global__ void k_deg_accum(const int* __restrict__ col, float* __restrict__ deg, int e) {
    int i = blockIdx.x * blockDim.x + threadIdx.x;
    if (i < e) unsafeAtomicAdd(&deg[col[i]], 1.0f);
}

__global__ void k_dinv(float* __restrict__ deg, int n) {
    int i = blockIdx.x * blockDim.x + threadIdx.x;
    if (i < n) {
        float d = deg[i];
        deg[i] = (d > 0.0f) ? __frsqrt_rn(fmaxf(d, 1.0f)) : 0.0f;
    }
}

// h[i][:] = emb[x[i]][:]   (16 threads per node, float4 each)
__global__ void k_embed(const int* __restrict__ x, const float* __restrict__ emb,
                        float* __restrict__ h) {
    int tid = blockIdx.x * blockDim.x + threadIdx.x;
    int i = tid >> 4, j = tid & 15;
    if (i >= N_NODES) return;
    const float4 v = *(const float4*)(emb + (size_t)x[i] * EMBED + j * 4);
    *(float4*)(h + (size_t)i * EMBED + j * 4) = v;
}

// ---------------- fp32 WMMA GEMM: O[nrows,64] = H[nrows,64] @ W[64,64] ----------------
// 256 threads = 8 waves; each wave computes a 32x64 output stripe with
// eight 16x16 f32 accumulators over 16 K-steps of V_WMMA_F32_16X16X4_F32.
// W is staged to LDS with gfx1250 async-to-LDS loads, then re-packed so each
// WMMA B fragment {W[k][n], W[k+1][n]} is one aligned ds_load_b64.

__global__ __launch_bounds__(256) void k_gemm64(const float* __restrict__ H,
                                                const float* __restrict__ W,
                                                float* __restrict__ O, int nrows) {
    __shared__ __align__(16) float Wraw[64 * 64];   // row-major staging
    __shared__ __align__(16) float Ws[64 * 64];     // k-pair swizzled: [kp][n][2]

    // async stage: 256 lanes x 4 passes x 16B = 16KB, tracked by ASYNCcnt
    for (int p = 0; p < 4; ++p) {
        const int idx = p * 1024 + threadIdx.x * 4;              // float index, 16B aligned
        unsigned lds  = (unsigned)(uintptr_t)&Wraw[idx];
        unsigned long long ga = (unsigned long long)(uintptr_t)(W + idx);
        asm volatile("global_load_async_to_lds_b128 %0, %1, off"
                     :: "v"(lds), "v"(ga) : "memory");
    }
    asm volatile("s_wait_asynccnt 0x0" ::: "memory");
    __syncthreads();

    // repack into k-pair-interleaved layout
    for (int i = threadIdx.x; i < 64 * 64; i += 256) {
        const int k = i >> 6, n = i & 63;
        Ws[(k >> 1) * 128 + n * 2 + (k & 1)] = Wraw[i];
    }
    __syncthreads();

    const int wave = threadIdx.x >> 5;
    const int lane = threadIdx.x & 31;
    const int hi   = lane >> 4;      // 0: K pair {4kk,4kk+1}, rows M=r ; 1: K pair {4kk+2,4kk+3}, rows M=r+8
    const int ln   = lane & 15;

    const int row0    = (blockIdx.x * 8 + wave) * 32;
    const bool active = (row0 < nrows);              // 32 | nrows -> all-or-nothing tiles
    int ar0 = row0 + ln;        if (ar0 >= nrows) ar0 = nrows - 1;
    int ar1 = row0 + 16 + ln;   if (ar1 >= nrows) ar1 = nrows - 1;
    const float* hp0 = H + (size_t)ar0 * 64;
    const float* hp1 = H + (size_t)ar1 * 64;
    const v2f* Wv = (const v2f*)Ws;                  // element (kp*64 + n)

    v8f c00 = {}, c01 = {}, c02 = {}, c03 = {};      // rows row0..row0+15
    v8f c10 = {}, c11 = {}, c12 = {}, c13 = {};      // rows row0+16..row0+31

    for (int kk = 0; kk < 16; ++kk) {
        const int kp = kk * 2 + hi;                  // k-pair index
        const v2f* wrow = Wv + kp * 64;
        const v2f b0 = wrow[ln +  0];
        const v2f b1 = wrow[ln + 16];
        const v2f b2 = wrow[ln + 32];
        const v2f b3 = wrow[ln + 48];
        const v2f a0 = *(const v2f*)(hp0 + kk * 4 + hi * 2);
        const v2f a1 = *(const v2f*)(hp1 + kk * 4 + hi * 2);

        c00 = __builtin_amdgcn_wmma_f32_16x16x4_f32(false, a0, false, b0, (short)0, c00, false, false);
        c01 = __builtin_amdgcn_wmma_f32_16x16x4_f32(false, a0, false, b1, (short)0, c01, false, false);
        c02 = __builtin_amdgcn_wmma_f32_16x16x4_f32(false, a0, false, b2, (short)0, c02, false, false);
        c03 = __builtin_amdgcn_wmma_f32_16x16x4_f32(false, a0, false, b3, (short)0, c03, false, false);
        c10 = __builtin_amdgcn_wmma_f32_16x16x4_f32(false, a1, false, b0, (short)0, c10, false, false);
        c11 = __builtin_amdgcn_wmma_f32_16x16x4_f32(false, a1, false, b1, (short)0, c11, false, false);
        c12 = __builtin_amdgcn_wmma_f32_16x16x4_f32(false, a1, false, b2, (short)0, c12, false, false);
        c13 = __builtin_amdgcn_wmma_f32_16x16x4_f32(false, a1, false, b3, (short)0, c13, false, false);
    }

    if (active) {
        const int m0 = row0 + hi * 8;
        for (int r = 0; r < 8; ++r) {
            float* op0 = O + (size_t)(m0 + r) * 64 + ln;
            op0[ 0] = c00[r];
            op0[16] = c01[r];
            op0[32] = c02[r];
            op0[48] = c03[r];
            float* op1 = O + (size_t)(m0 + 16 + r) * 64 + ln;
            op1[ 0] = c10[r];
            op1[16] = c11[r];
            op1[32] = c12[r];
            op1[48] = c13[r];
        }
    }
}

// ---------------- edge scatter: aggr[col] += hW[row] * dinv[row]*dinv[col] ----------------
// 16 threads per edge, float4 per thread.

__global__ void k_edge_scatter(const int* __restrict__ row, const int* __restrict__ col,
                               const float* __restrict__ dinv, const float* __restrict__ hW,
                               float* __restrict__ aggr, int e) {
    int tid = blockIdx.x * blockDim.x + threadIdx.x;
    int ei = tid >> 4, j = tid & 15;
    if (ei >= e) return;
    const int r = row[ei], c = col[ei];
    const float nrm = dinv[r] * dinv[c];
    const float4 v = *(const float4*)(hW + (size_t)r * HID + j * 4);
    float* dst = aggr + (size_t)c * HID + j * 4;
    unsafeAtomicAdd(dst + 0, v.x * nrm);
    unsafeAtomicAdd(dst + 1, v.y * nrm);
    unsafeAtomicAdd(dst + 2, v.z * nrm);
    unsafeAtomicAdd(dst + 3, v.w * nrm);
}

// ---------------- epilogue: out = relu(aggr + hW*dinv^2 + b) ----------------

__global__ void k_post(const float* __restrict__ aggr, const float* __restrict__ hW,
                       const float* __restrict__ dinv, const float* __restrict__ b,
                       float* __restrict__ out) {
    int tid = blockIdx.x * blockDim.x + threadIdx.x;
    int i = tid >> 6, f = tid & 63;
    if (i >= N_NODES) return;
    const float di = dinv[i];
    const float v = aggr[(size_t)i * HID + f] + hW[(size_t)i * HID + f] * di * di + b[f];
    out[(size_t)i * HID + f] = fmaxf(v, 0.0f);
}

// ---------------- pooling: per-graph sums + counts ----------------

__global__ void k_pool(const float* __restrict__ h, const int* __restrict__ batch,
                       float* __restrict__ gsum, float* __restrict__ gcnt) {
    int tid = blockIdx.x * blockDim.x + threadIdx.x;
    int i = tid >> 4, j = tid & 15;
    if (i >= N_NODES) return;
    const int g = batch[i];
    const float4 v = *(const float4*)(h + (size_t)i * HID + j * 4);
    float* dst = gsum + (size_t)g * HID + j * 4;
    unsafeAtomicAdd(dst + 0, v.x);
    unsafeAtomicAdd(dst + 1, v.y);
    unsafeAtomicAdd(dst + 2, v.z);
    unsafeAtomicAdd(dst + 3, v.w);
    if (j == 0) unsafeAtomicAdd(&gcnt[g], 1.0f);
}

// ---------------- head: out[g,c] = (gsum[g]/cnt) @ Wl + bl ----------------

__global__ void k_head(const float* __restrict__ gsum, const float* __restrict__ gcnt,
                       const float* __restrict__ Wl, const float* __restrict__ bl,
                       float* __restrict__ out) {
    int t = blockIdx.x * blockDim.x + threadIdx.x;
    if (t >= N_GRAPHS * N_CL) return;
    const int g = t / N_CL, c = t % N_CL;
    const float inv = 1.0f / fmaxf(gcnt[g], 1.0f);
    float s = 0.0f;
    for (int f = 0; f < HID; ++f)
        s += gsum[(size_t)g * HID + f] * Wl[f * N_CL + c];
    out[t] = s * inv + bl[c];
}

// ---------------- driver ----------------

extern "C" void kernel_launch(void* const* d_in, const int* in_sizes, int n_in,
                              void* d_out, int out_size, void* d_ws, size_t ws_size,
                              hipStream_t stream) {
    (void)in_sizes; (void)n_in; (void)out_size; (void)ws_size;
    const int*   x    = (const int*)d_in[0];
    const int*   erow = (const int*)d_in[1];             // edge_index[0]
    const int*   ecol = erow + N_EDGES;                  // edge_index[1]
    const int*   batch= (const int*)d_in[2];
    const float* emb  = (const float*)d_in[3];
    const float* W1   = (const float*)d_in[4];
    const float* b1   = (const float*)d_in[5];
    const float* W2   = (const float*)d_in[6];
    const float* b2   = (const float*)d_in[7];
    const float* Wl   = (const float*)d_in[8];
    const float* bl   = (const float*)d_in[9];
    float* out = (float*)d_out;

    // workspace carve-up
    float* deg  = (float*)d_ws;                          // N (becomes dinv)
    float* hA   = deg  + N_NODES;                        // N*64
    float* hB   = hA   + (size_t)N_NODES * HID;          // N*64
    float* hC   = hB   + (size_t)N_NODES * HID;          // N*64
    float* gsum = hC   + (size_t)N_NODES * HID;          // G*64
    float* gcnt = gsum + (size_t)N_GRAPHS * HID;         // G

    const int T = 256;
    const int nfeat = N_NODES * HID;

    // degrees (self loops fold in as the initial 1.0) -> dinv
    k_fill_f32<<<cdiv(N_NODES, T), T, 0, stream>>>(deg, 1.0f, N_NODES);
    k_deg_accum<<<cdiv(N_EDGES, T), T, 0, stream>>>(ecol, deg, N_EDGES);
    k_dinv<<<cdiv(N_NODES, T), T, 0, stream>>>(deg, N_NODES);

    // h0 = emb[x]
    k_embed<<<cdiv(N_NODES * 16, T), T, 0, stream>>>(x, emb, hA);

    // ---- layer 1 ----
    k_gemm64<<<cdiv(N_NODES, 256), 256, 0, stream>>>(hA, W1, hB, N_NODES);
    k_fill_f32<<<cdiv(nfeat, T), T, 0, stream>>>(hC, 0.0f, nfeat);
    k_edge_scatter<<<cdiv(N_EDGES * 16, T), T, 0, stream>>>(erow, ecol, deg, hB, hC, N_EDGES);
    k_post<<<cdiv(nfeat, T), T, 0, stream>>>(hC, hB, deg, b1, hA);

    // ---- layer 2 ----
    k_gemm64<<<cdiv(N_NODES, 256), 256, 0, stream>>>(hA, W2, hB, N_NODES);
    k_fill_f32<<<cdiv(nfeat, T), T, 0, stream>>>(hC, 0.0f, nfeat);
    k_edge_scatter<<<cdiv(N_EDGES * 16, T), T, 0, stream>>>(erow, ecol, deg, hB, hC, N_EDGES);
    k_post<<<cdiv(nfeat, T), T, 0, stream>>>(hC, hB, deg, b2, hA);

    // ---- global mean pool + head ----
    const int npool = N_GRAPHS * HID + N_GRAPHS;
    k_fill_f32<<<cdiv(npool, T), T, 0, stream>>>(gsum, 0.0f, npool);
    k_pool<<<cdiv(N_NODES * 16, T), T, 0, stream>>>(hA, batch, gsum, gcnt);
    k_head<<<cdiv(N_GRAPHS * N_CL, T), T, 0, stream>>>(gsum, gcnt, Wl, bl, out);
}